// NeuralTSBandit_2516850835967
// MI455X (gfx1250) — compile-verified
//
#include <hip/hip_runtime.h>
#include <math.h>

// Problem constants (match reference)
#define D_IN   128
#define H_DIM  48
#define B_SZ   32
#define P_DIM  6241          // D*H + H + H + 1
#define EPS_REG 1e-4f
#define NITER  80            // CG: <=65 distinct eigenvalues => converged well before this
#define PAD_P  6256          // P rounded up to multiple of 16

typedef float v2f __attribute__((ext_vector_type(2)));
typedef float v8f __attribute__((ext_vector_type(8)));

// ---------------------------------------------------------------------------
// Kernel 1: MLP forward (WMMA f32 16x16x4) + backward grads + CG init.
// One workgroup, 256 threads (8 wave32s). Waves 0..5 each own one 16x16 tile
// of pre = emb[32x128] @ W1^T[128x48].
// ---------------------------------------------------------------------------
__launch_bounds__(256)
__global__ void nts_setup_kernel(const float* __restrict__ emb,
                                 const float* __restrict__ W1,
                                 const float* __restrict__ b1,
                                 const float* __restrict__ W2,
                                 const float* __restrict__ b2,
                                 float* __restrict__ out_mu,   // d_out[0..31]
                                 float* __restrict__ g,
                                 float* __restrict__ x,
                                 float* __restrict__ r,
                                 float* __restrict__ p,
                                 float* __restrict__ rs)       // rs[0] = g.g
{
    __shared__ float pre[B_SZ * H_DIM];   // 32x48 pre-activations (incl. b1)
    __shared__ float red[256];

    const int tid  = threadIdx.x;
    const int wave = tid >> 5;
    const int lane = tid & 31;

    // ---- forward GEMM with v_wmma_f32_16x16x4_f32 -------------------------
    if (wave < 6) {
        const int mt   = wave / 3;        // batch tile: 0..1
        const int nt   = wave % 3;        // hidden tile: 0..2
        const int half = lane >> 4;       // 0: lanes 0-15, 1: lanes 16-31
        const int l16  = lane & 15;
        const int m    = mt * 16 + l16;   // A-matrix row (batch)
        const int n    = nt * 16 + l16;   // B-matrix col (hidden unit)
        v8f c = {};
        #pragma unroll 8
        for (int kk = 0; kk < D_IN / 4; ++kk) {
            // ISA 7.12.2: f32 A 16x4 -> lanes 0-15 hold K0,K1; lanes 16-31 K2,K3
            const int kb = kk * 4 + half * 2;
            v2f a, b;
            a.x = emb[m * D_IN + kb];
            a.y = emb[m * D_IN + kb + 1];
            // B 4x16 (row = K striped across lanes): VGPR j holds K=j (lo) / K=j+2 (hi)
            b.x = W1[n * D_IN + kb];
            b.y = W1[n * D_IN + kb + 1];
            c = __builtin_amdgcn_wmma_f32_16x16x4_f32(false, a, false, b,
                                                      (short)0, c, false, false);
        }
        // C/D layout: VGPR rr -> M = rr + half*8, N = l16
        #pragma unroll
        for (int rr = 0; rr < 8; ++rr) {
            const int mm = mt * 16 + rr + half * 8;
            pre[mm * H_DIM + n] = c[rr] + b1[n];
        }
    }
    __syncthreads();

    // ---- mu = relu(pre) @ W2^T + b2 --------------------------------------
    if (tid < B_SZ) {
        float mu = b2[0];
        #pragma unroll
        for (int h = 0; h < H_DIM; ++h) {
            float a = pre[tid * H_DIM + h];
            mu = fmaf(fmaxf(a, 0.0f), W2[h], mu);
        }
        out_mu[tid] = mu;
    }

    // ---- gradients + CG init (g, r=p=g, x=0) -----------------------------
    float ss = 0.0f;   // local partial of g.g

    // grad W1 [48x128] -> g[0..6143]; dpre[i,h] = (pre>0) * W2[h]
    for (int idx = tid; idx < H_DIM * D_IN; idx += 256) {
        const int h = idx >> 7;        // idx / 128
        const int d = idx & 127;       // idx % 128
        const float w2h = W2[h];
        float s = 0.0f;
        #pragma unroll
        for (int i = 0; i < B_SZ; ++i) {
            const float mask = (pre[i * H_DIM + h] > 0.0f) ? w2h : 0.0f;
            s = fmaf(mask, emb[i * D_IN + d], s);
        }
        g[idx] = s; r[idx] = s; p[idx] = s; x[idx] = 0.0f;
        ss = fmaf(s, s, ss);
    }
    // grad b1 -> g[6144..6191]; grad W2 -> g[6192..6239]
    if (tid < H_DIM) {
        const int h = tid;
        float sb1 = 0.0f, sw2 = 0.0f;
        #pragma unroll
        for (int i = 0; i < B_SZ; ++i) {
            const float a = pre[i * H_DIM + h];
            if (a > 0.0f) { sb1 += W2[h]; sw2 += a; }
        }
        int i1 = H_DIM * D_IN + h;
        g[i1] = sb1; r[i1] = sb1; p[i1] = sb1; x[i1] = 0.0f;
        int i2 = H_DIM * D_IN + H_DIM + h;
        g[i2] = sw2; r[i2] = sw2; p[i2] = sw2; x[i2] = 0.0f;
        ss = fmaf(sb1, sb1, fmaf(sw2, sw2, ss));
    }
    // grad b2 = B -> g[6240]
    if (tid == 0) {
        const float v = (float)B_SZ;
        const int i3 = P_DIM - 1;
        g[i3] = v; r[i3] = v; p[i3] = v; x[i3] = 0.0f;
        ss = fmaf(v, v, ss);
    }

    // deterministic block reduction for rs[0] = g.g
    red[tid] = ss;
    __syncthreads();
    for (int st = 128; st > 0; st >>= 1) {
        if (tid < st) red[tid] += red[tid + st];
        __syncthreads();
    }
    if (tid == 0) rs[0] = red[0];
}

// ---------------------------------------------------------------------------
// Kernel 2: q = (U + eps*I) p  (bandwidth bound; 156MB/pass, resident in 192MB L2)
// 256 threads = 8 waves; 2 rows per wave, p staged in LDS, coalesced b32 streams.
// Also emits per-block partial of p.q (no atomics -> deterministic).
// ---------------------------------------------------------------------------
__launch_bounds__(256)
__global__ void nts_matvec_kernel(const float* __restrict__ U,
                                  const float* __restrict__ p,
                                  float* __restrict__ q,
                                  float* __restrict__ pqpart)
{
    __shared__ float pl[PAD_P];
    __shared__ float wred[8];

    const int tid  = threadIdx.x;
    const int lane = tid & 31;
    const int wave = tid >> 5;

    for (int i = tid; i < P_DIM; i += 256) pl[i] = p[i];
    __syncthreads();

    const int r0 = blockIdx.x * 16 + wave * 2;
    const int r1 = r0 + 1;
    float acc0 = 0.0f, acc1 = 0.0f;

    if (r1 < P_DIM) {                       // common path: two rows per wave
        const float* __restrict__ row0 = U + (size_t)r0 * P_DIM;
        const float* __restrict__ row1 = U + (size_t)r1 * P_DIM;
        #pragma unroll 8
        for (int j = lane; j < P_DIM; j += 32) {
            const float pv = pl[j];
            acc0 = fmaf(row0[j], pv, acc0);
            acc1 = fmaf(row1[j], pv, acc1);
        }
    } else if (r0 < P_DIM) {                // tail row
        const float* __restrict__ row0 = U + (size_t)r0 * P_DIM;
        #pragma unroll 8
        for (int j = lane; j < P_DIM; j += 32) {
            acc0 = fmaf(row0[j], pl[j], acc0);
        }
    }

    // wave32 reductions
    for (int off = 16; off > 0; off >>= 1) {
        acc0 += __shfl_down(acc0, off, 32);
        acc1 += __shfl_down(acc1, off, 32);
    }

    if (lane == 0) {
        float part = 0.0f;
        if (r0 < P_DIM) {
            const float q0 = acc0 + EPS_REG * pl[r0];
            q[r0] = q0; part = fmaf(pl[r0], q0, part);
        }
        if (r1 < P_DIM) {
            const float q1 = acc1 + EPS_REG * pl[r1];
            q[r1] = q1; part = fmaf(pl[r1], q1, part);
        }
        wred[wave] = part;
    }
    __syncthreads();
    if (tid == 0) {
        float s = 0.0f;
        #pragma unroll
        for (int w = 0; w < 8; ++w) s += wred[w];
        pqpart[blockIdx.x] = s;
    }
}

// ---------------------------------------------------------------------------
// Kernel 3: alpha = rs[t]/(p.q); x += alpha p; r -= alpha q; partials of r.r
// Every block redundantly reduces pqpart in a fixed order (deterministic).
// ---------------------------------------------------------------------------
__launch_bounds__(256)
__global__ void nts_cg_update1(const float* __restrict__ pqpart, int npq,
                               const float* __restrict__ rs, int t,
                               const float* __restrict__ p,
                               const float* __restrict__ q,
                               float* __restrict__ x,
                               float* __restrict__ r,
                               float* __restrict__ rspart)
{
    __shared__ float red[256];
    const int tid = threadIdx.x;

    float s = 0.0f;
    for (int i = tid; i < npq; i += 256) s += pqpart[i];
    red[tid] = s;
    __syncthreads();
    for (int st = 128; st > 0; st >>= 1) {
        if (tid < st) red[tid] += red[tid + st];
        __syncthreads();
    }
    const float alpha = rs[t] / red[0];
    __syncthreads();

    const int i = blockIdx.x * 256 + tid;
    float rr = 0.0f;
    if (i < P_DIM) {
        x[i] = fmaf(alpha, p[i], x[i]);
        const float rn = fmaf(-alpha, q[i], r[i]);
        r[i] = rn;
        rr = rn * rn;
    }
    red[tid] = rr;
    __syncthreads();
    for (int st = 128; st > 0; st >>= 1) {
        if (tid < st) red[tid] += red[tid + st];
        __syncthreads();
    }
    if (tid == 0) rspart[blockIdx.x] = red[0];
}

// ---------------------------------------------------------------------------
// Kernel 4: beta = rs[t+1]/rs[t]; p = r + beta p; publish rs[t+1]
// ---------------------------------------------------------------------------
__launch_bounds__(256)
__global__ void nts_cg_update2(const float* __restrict__ rspart, int nrs,
                               float* __restrict__ rs, int t,
                               const float* __restrict__ r,
                               float* __restrict__ p)
{
    __shared__ float red[256];
    const int tid = threadIdx.x;
    red[tid] = (tid < nrs) ? rspart[tid] : 0.0f;
    __syncthreads();
    for (int st = 128; st > 0; st >>= 1) {
        if (tid < st) red[tid] += red[tid + st];
        __syncthreads();
    }
    const float rs_new = red[0];
    const float beta   = rs_new / rs[t];

    const int i = blockIdx.x * 256 + tid;
    if (i < P_DIM) p[i] = fmaf(beta, p[i], r[i]);
    if (blockIdx.x == 0 && tid == 0) rs[t + 1] = rs_new;
}

// ---------------------------------------------------------------------------
// Kernel 5: sigma = sqrt(max(g.x / H, 1e-6))  (lambda = nu = 1)
// ---------------------------------------------------------------------------
__launch_bounds__(256)
__global__ void nts_final_kernel(const float* __restrict__ g,
                                 const float* __restrict__ x,
                                 float* __restrict__ out_sigma)  // d_out + 32
{
    __shared__ float red[256];
    const int tid = threadIdx.x;
    float s = 0.0f;
    for (int i = tid; i < P_DIM; i += 256) s = fmaf(g[i], x[i], s);
    red[tid] = s;
    __syncthreads();
    for (int st = 128; st > 0; st >>= 1) {
        if (tid < st) red[tid] += red[tid + st];
        __syncthreads();
    }
    if (tid == 0) {
        const float sig2 = red[0] / (float)H_DIM;
        out_sigma[0] = sqrtf(fmaxf(sig2, 1e-6f));
    }
}

// ---------------------------------------------------------------------------
extern "C" void kernel_launch(void* const* d_in, const int* in_sizes, int n_in,
                              void* d_out, int out_size, void* d_ws, size_t ws_size,
                              hipStream_t stream) {
    const float* emb = (const float*)d_in[0];   // [32,128]
    const float* W1  = (const float*)d_in[1];   // [48,128]
    const float* b1  = (const float*)d_in[2];   // [48]
    const float* W2  = (const float*)d_in[3];   // [1,48]
    const float* b2  = (const float*)d_in[4];   // [1]
    const float* U   = (const float*)d_in[5];   // [6241,6241]
    float* out = (float*)d_out;                 // [0..31]=mu, [32]=sigma

    float* ws = (float*)d_ws;
    float* g      = ws;                  // PAD_P
    float* x      = g + PAD_P;           // PAD_P
    float* r      = x + PAD_P;           // PAD_P
    float* p      = r + PAD_P;           // PAD_P
    float* q      = p + PAD_P;           // PAD_P
    float* rs     = q + PAD_P;           // NITER+1 (pad 96)
    float* pqpart = rs + 96;             // NB1 slots
    const int NB1 = (P_DIM + 15) / 16;   // 391 matvec blocks (16 rows each)
    const int NB2 = (P_DIM + 255) / 256; // 25 vector-update blocks
    float* rspart = pqpart + ((NB1 + 15) & ~15);

    nts_setup_kernel<<<1, 256, 0, stream>>>(emb, W1, b1, W2, b2, out,
                                            g, x, r, p, rs);

    for (int t = 0; t < NITER; ++t) {
        nts_matvec_kernel<<<NB1, 256, 0, stream>>>(U, p, q, pqpart);
        nts_cg_update1<<<NB2, 256, 0, stream>>>(pqpart, NB1, rs, t, p, q, x, r, rspart);
        nts_cg_update2<<<NB2, 256, 0, stream>>>(rspart, NB2, rs, t, r, p);
    }

    nts_final_kernel<<<1, 256, 0, stream>>>(g, x, out + B_SZ);
}